// KNN_49177375539581
// MI455X (gfx1250) — compile-verified
//
#include <hip/hip_runtime.h>
#include <float.h>
#include <math.h>

// Problem constants (match the JAX reference exactly).
#define NBATCH 8
#define NREF   8192
#define NQ     2048
#define KNN    32
#define QTILE  16                 // queries per wave (WMMA M)
#define RTILE  16                 // refs per WMMA (WMMA N)
#define NCHUNK (NREF / RTILE)     // 512

typedef __attribute__((ext_vector_type(2))) float v2f;
typedef __attribute__((ext_vector_type(8))) float v8f;

// One wave (32 threads) per (batch, 16-query tile).
//  - ref chunks staged LDS-side with global_load_async_to_lds_b32 (double buffer,
//    counted s_wait_asynccnt for fetch/compute overlap)
//  - exact d^2 via V_WMMA_F32_16X16X4_F32:  [qx qy qz 1] x [-2r | ||r||^2] + C(||q||^2)
//  - ballot-based fast reject: skip tile scatter + selection when no candidate
//    beats its row's current 32nd-smallest distance
__global__ __launch_bounds__(32) void knn_wmma_kernel(
    const float* __restrict__ ref,
    const float* __restrict__ query,
    float* __restrict__ outD,
    int*   __restrict__ outI) {

    __shared__ float tile[QTILE * RTILE];   // d^2 tile, row-major [query][ref]
    __shared__ float topd[QTILE][KNN];      // current best-32 distances^2
    __shared__ int   topi[QTILE][KNN];      // current best-32 indices
    __shared__ float qn[QTILE];             // ||q||^2 per query row
    __shared__ float rowmax[QTILE];         // per-row current 32nd-smallest d^2
    __shared__ float rbuf[2][64];           // double-buffered ref chunk (48 DW used)

    const int lane  = threadIdx.x;                 // 0..31 (wave32)
    const int b     = blockIdx.x / (NQ / QTILE);
    const int qt    = blockIdx.x % (NQ / QTILE);
    const int qbase = qt * QTILE;

    const int mrow  = lane & 15;          // row (query) for A, col (ref) for B
    const int kb    = (lane >> 4) * 2;    // K base: lanes 0-15 -> K=0,1 ; 16-31 -> K=2,3
    const int col   = lane & 15;
    const int rbase = (lane >> 4) * 8;    // D-matrix: row = v + rbase

    // ---- A matrix: 16 queries x [x, y, z, 1] (16x4 f32, 2 VGPRs) ----
    const float* qp = query + ((size_t)b * NQ + qbase + mrow) * 3;
    const float qx = qp[0], qy = qp[1], qz = qp[2];
    v2f a;
    a.x = (kb == 0) ? qx : qz;
    a.y = (kb == 0) ? qy : 1.0f;

    if (lane < QTILE) {
        qn[lane]     = qx * qx + qy * qy + qz * qz;
        rowmax[lane] = FLT_MAX;
        for (int i = 0; i < KNN; ++i) { topd[lane][i] = FLT_MAX; topi[lane][i] = -1; }
    }
    __syncthreads();

    // ---- C accumulator: ||q||^2 broadcast along rows of the 16x16 tile ----
    v8f cbase;
#pragma unroll
    for (int v = 0; v < 8; ++v) cbase[v] = qn[rbase + v];

    const float* refb = ref + (size_t)b * NREF * 3;

    // ---- async stage of one 16-point ref chunk (48 DWORDs) into rbuf[buf] ----
    auto stage = [&](int c, int buf) {
        const int base = c * RTILE * 3;                 // dword index of chunk start
        const float* s0 = refb + base + lane;           // dwords 0..31
        int gi = base + 32 + lane;                      // dwords 32..63 (48+ unused)
        if (gi > NREF * 3 - 1) gi = NREF * 3 - 1;       // clamp tail overrun
        const float* s1 = refb + gi;
        const unsigned l0 = (unsigned)(uintptr_t)(const void*)&rbuf[buf][lane];
        const unsigned l1 = (unsigned)(uintptr_t)(const void*)&rbuf[buf][32 + lane];
        asm volatile("global_load_async_to_lds_b32 %0, %1, off"
                     :: "v"(l0), "v"(s0) : "memory");
        asm volatile("global_load_async_to_lds_b32 %0, %1, off"
                     :: "v"(l1), "v"(s1) : "memory");
    };

    stage(0, 0);                       // prologue: chunk 0 in flight

    float curmax = FLT_MAX;            // current 32nd-smallest (lanes 0-15 only)
    int   maxpos = 0;

    for (int c = 0; c < NCHUNK; ++c) {
        const int buf = c & 1;
        const int n0  = c * RTILE;

        if (c + 1 < NCHUNK) {
            stage(c + 1, buf ^ 1);                       // prefetch next chunk
            asm volatile("s_wait_asynccnt 0x2" ::: "memory");  // current chunk done
        } else {
            asm volatile("s_wait_asynccnt 0x0" ::: "memory");
        }

        // ---- B matrix from LDS: col n = [-2rx, -2ry, -2rz, ||r||^2] ----
        const float rx = rbuf[buf][col * 3 + 0];
        const float ry = rbuf[buf][col * 3 + 1];
        const float rz = rbuf[buf][col * 3 + 2];
        v2f bm;
        if (kb == 0) { bm.x = -2.0f * rx; bm.y = -2.0f * ry; }
        else         { bm.x = -2.0f * rz; bm.y = rx * rx + ry * ry + rz * rz; }

        // D = A x B + C  ->  exact squared Euclidean distances, 16x16 tile.
        v8f d2 = __builtin_amdgcn_wmma_f32_16x16x4_f32(
            false, a, false, bm, (short)0, cbase, false, false);

        // ---- fast reject: does any candidate beat its row's threshold? ----
        bool cand = false;
#pragma unroll
        for (int v = 0; v < 8; ++v) cand |= (d2[v] < rowmax[rbase + v]);

        if (__builtin_amdgcn_ballot_w32(cand)) {
            // Scatter tile to LDS: lane holds (row = v + rbase, col).
#pragma unroll
            for (int v = 0; v < 8; ++v)
                tile[(rbase + v) * RTILE + col] = d2[v];
            __syncthreads();

            // ---- top-32 maintenance: lane q owns query q ----
            if (lane < QTILE) {
                for (int j = 0; j < RTILE; ++j) {
                    const float d = tile[lane * RTILE + j];
                    if (d < curmax) {
                        topd[lane][maxpos] = d;
                        topi[lane][maxpos] = n0 + j;
                        float m = -FLT_MAX; int mp = 0;
                        for (int i = 0; i < KNN; ++i) {
                            const float t = topd[lane][i];
                            if (t > m) { m = t; mp = i; }
                        }
                        curmax = m; maxpos = mp;
                    }
                }
                rowmax[lane] = curmax;
            }
            __syncthreads();
        }
    }

    // ---- final ascending sort of the 32 survivors + output ----
    if (lane < QTILE) {
        for (int i = 0; i < KNN - 1; ++i) {
            float mn = topd[lane][i]; int mp = i;
            for (int j = i + 1; j < KNN; ++j) {
                const float t = topd[lane][j];
                if (t < mn) { mn = t; mp = j; }
            }
            if (mp != i) {
                const float td = topd[lane][i]; topd[lane][i] = topd[lane][mp]; topd[lane][mp] = td;
                const int   ti = topi[lane][i]; topi[lane][i] = topi[lane][mp]; topi[lane][mp] = ti;
            }
        }
        const size_t obase = ((size_t)b * NQ + qbase + lane) * KNN;
        for (int i = 0; i < KNN; ++i) {
            const float d2v = topd[lane][i];
            outD[obase + i] = sqrtf(fmaxf(d2v, 0.0f));
            outI[obase + i] = topi[lane][i];
        }
    }
}

extern "C" void kernel_launch(void* const* d_in, const int* in_sizes, int n_in,
                              void* d_out, int out_size, void* d_ws, size_t ws_size,
                              hipStream_t stream) {
    (void)in_sizes; (void)n_in; (void)out_size; (void)d_ws; (void)ws_size;

    const float* ref   = (const float*)d_in[0];   // [8, 8192, 3] f32
    const float* query = (const float*)d_in[1];   // [8, 2048, 3] f32

    // d_out = D [8,2048,32] f32  ||  I [8,2048,32] i32 (bit-stored after D)
    float* outD = (float*)d_out;
    int*   outI = (int*)((float*)d_out + (size_t)NBATCH * NQ * KNN);

    const int nblocks = NBATCH * (NQ / QTILE);    // 1024 single-wave blocks
    knn_wmma_kernel<<<nblocks, 32, 0, stream>>>(ref, query, outD, outI);
}